// SupConLoss_79757542686729
// MI455X (gfx1250) — compile-verified
//
#include <hip/hip_runtime.h>

// ---- constants from the reference ----
#define BTOT   4096      // batch
#define SPLIT  256       // spatial_size^2
#define MBLK   16        // BTOT / SPLIT
#define KF     384       // feature dim of modeloutput_f / Pool_ag
#define KS     70        // dim of modeloutput_s_pr / Pool_sp
#define KZ     128       // dim of modeloutput_z / z_mix
#define NPOOL  2048      // pool rows
#define NSPL   4         // column splits per row tile (partials are additive)
#define TEMP_F      0.07f
#define BASE_TEMP_F 0.07f
#define RHO_F       0.3f
#define LMBD_F      0.5f

typedef __attribute__((ext_vector_type(2))) float v2f;
typedef __attribute__((ext_vector_type(8))) float v8f;

// One 16x16 tile of X * Y^T with fp32 WMMA (V_WMMA_F32_16X16X4_F32).
// A side comes from LDS (row tile staged once per block, padded stride KP,
// zero-filled beyond K) -> ds_load_b64. B side streams from global with a
// 2x8-fragment register double buffer so loads of chunk c+1 are independent
// of the WMMAs consuming chunk c (breaks the load->wait(0)->wmma serial chain).
template<int K>
__device__ inline v8f mm_tile(const float* Al, const float* __restrict__ Bg,
                              int brow0) {
  constexpr int KP    = (K + 3) & ~3;
  constexpr int STEPS = KP / 4;
  constexpr int CH    = 8;
  constexpr int NCH   = (STEPS + CH - 1) / CH;
  const int lane = threadIdx.x & 31;
  const int kb = (lane >> 4) << 1;                 // 0 or 2
  const float* pa = Al + (lane & 15) * KP + kb;    // LDS, zero-padded
  const float* pb = Bg + (long)(brow0 + (lane & 15)) * K + kb;
  v2f bb[2][CH];
  v8f acc = {0.f, 0.f, 0.f, 0.f, 0.f, 0.f, 0.f, 0.f};

#pragma unroll
  for (int s = 0; s < CH; ++s) {                   // prime buffer 0
    if (s < STEPS) {
      const int k = s * 4;
      if ((K & 3) != 0 && s == STEPS - 1) {        // guarded tail (K=70)
        const int kk = k + kb;
        bb[0][s].x = (kk     < K) ? pb[k]     : 0.f;
        bb[0][s].y = (kk + 1 < K) ? pb[k + 1] : 0.f;
      } else {
        bb[0][s].x = pb[k];
        bb[0][s].y = pb[k + 1];
      }
    }
  }

#pragma unroll
  for (int c = 0; c < NCH; ++c) {                  // fully unrolled: cur/nxt
    const int cur = c & 1, nxt = cur ^ 1;          // are compile-time consts
    if (c + 1 < NCH) {
#pragma unroll
      for (int s = 0; s < CH; ++s) {
        const int step = (c + 1) * CH + s;
        if (step < STEPS) {
          const int k = step * 4;
          if ((K & 3) != 0 && step == STEPS - 1) {
            const int kk = k + kb;
            bb[nxt][s].x = (kk     < K) ? pb[k]     : 0.f;
            bb[nxt][s].y = (kk + 1 < K) ? pb[k + 1] : 0.f;
          } else {
            bb[nxt][s].x = pb[k];
            bb[nxt][s].y = pb[k + 1];
          }
        }
      }
    }
#pragma unroll
    for (int s = 0; s < CH; ++s) {
      const int step = c * CH + s;
      if (step < STEPS) {
        v2f a;
        a.x = pa[step * 4];
        a.y = pa[step * 4 + 1];
        acc = __builtin_amdgcn_wmma_f32_16x16x4_f32(false, a, false, bb[cur][s],
                                                    (short)0, acc, false, false);
      }
    }
  }
  return acc;
}

// R[i] = max_j (X @ P^T)[i,j].  4 waves per 16-row tile split the pool column
// tiles; per-wave shuffle max reduce, LDS max-combine across waves.
template<int K>
__global__ void __launch_bounds__(128)
rowmax_kernel(const float* __restrict__ X, const float* __restrict__ P,
              float* __restrict__ R, int ntiles) {
  constexpr int KP = (K + 3) & ~3;
  __shared__ float AT[16 * KP];
  __shared__ float redm[4][16];
  const int i = blockIdx.x;
  const int wave = threadIdx.x >> 5;
  const int lane = threadIdx.x & 31;
  const int h = lane >> 4;

  for (int idx = threadIdx.x; idx < 16 * KP; idx += 128) {
    const int r = idx / KP, c = idx % KP;
    AT[idx] = (c < K) ? X[(long)(i * 16 + r) * K + c] : 0.f;
  }
  __syncthreads();

  float rm[8];
#pragma unroll
  for (int v = 0; v < 8; ++v) rm[v] = -3.4e38f;
  for (int jt = wave; jt < ntiles; jt += 4) {
    v8f acc = mm_tile<K>(AT, P, jt * 16);
#pragma unroll
    for (int v = 0; v < 8; ++v) rm[v] = fmaxf(rm[v], acc[v]);
  }
#pragma unroll
  for (int v = 0; v < 8; ++v) {
#pragma unroll
    for (int m = 8; m >= 1; m >>= 1)
      rm[v] = fmaxf(rm[v], __shfl_xor(rm[v], m, 32));
  }
  if ((lane & 15) == 0) {
#pragma unroll
    for (int v = 0; v < 8; ++v) redm[wave][v + 8 * h] = rm[v];
  }
  __syncthreads();
  if (threadIdx.x < 16) {
    float m = redm[0][threadIdx.x];
#pragma unroll
    for (int w = 1; w < 4; ++w) m = fmaxf(m, redm[w][threadIdx.x]);
    R[i * 16 + threadIdx.x] = m;
  }
}

// Fused main pass.  Block = (row tile i, column split sp); 8 waves split the
// 64 column tiles of the split.  Row tile of all four matrices staged in LDS.
// All six per-row accumulators are additive over columns (no rowmax needed
// since |logits| <= 1/TEMP), so splits just sum partials into scratch.
__global__ void __launch_bounds__(256)
supcon_main(const float* __restrict__ f,  const float* __restrict__ spr,
            const float* __restrict__ z,  const float* __restrict__ zmix,
            const float* __restrict__ U,  const float* __restrict__ Rpoint,
            const float* __restrict__ Rema, float* __restrict__ part) {
  constexpr int KSP = (KS + 3) & ~3;
  constexpr int TPS = (BTOT / 16) / NSPL;          // tiles per split
  __shared__ float Af[16 * KF];
  __shared__ float Asp[16 * KSP];
  __shared__ float Az[16 * KZ];
  __shared__ float Am[16 * KZ];
  __shared__ float red[8][16][6];

  const int i    = blockIdx.x >> 2;
  const int sp   = blockIdx.x & (NSPL - 1);
  const int wave = threadIdx.x >> 5;
  const int lane = threadIdx.x & 31;
  const int h    = lane >> 4;
  const float invT = 1.0f / TEMP_F;

  for (int idx = threadIdx.x; idx < 16 * KF; idx += 256)
    Af[idx] = f[(long)(i * 16 + (idx / KF)) * KF + (idx % KF)];
  for (int idx = threadIdx.x; idx < 16 * KSP; idx += 256) {
    const int c = idx % KSP;
    Asp[idx] = (c < KS) ? spr[(long)(i * 16 + (idx / KSP)) * KS + c] : 0.f;
  }
  for (int idx = threadIdx.x; idx < 16 * KZ; idx += 256) {
    Az[idx] = z[(long)(i * 16 + (idx / KZ)) * KZ + (idx % KZ)];
    Am[idx] = zmix[(long)(i * 16 + (idx / KZ)) * KZ + (idx % KZ)];
  }
  __syncthreads();

  float rr[8], rre[8];
  int rowg[8], srow[8];
#pragma unroll
  for (int v = 0; v < 8; ++v) {
    rowg[v] = i * 16 + v + 8 * h;
    srow[v] = rowg[v] & (SPLIT - 1);
    rr[v]  = Rpoint[rowg[v]];
    rre[v] = Rema[rowg[v]];
  }

  float msum[8], wsum[8], swz[8], swm[8], dz[8], dm[8];
#pragma unroll
  for (int v = 0; v < 8; ++v)
    msum[v] = wsum[v] = swz[v] = swm[v] = dz[v] = dm[v] = 0.f;

  for (int jt = sp * TPS + wave; jt < (sp + 1) * TPS; jt += 8) {
    if (jt + 8 < (sp + 1) * TPS) {                 // pull next tile toward WGP
      const int rn = (jt + 8) * 16 + (lane & 15);
      __builtin_prefetch(f    + (long)rn * KF, 0, 3);
      __builtin_prefetch(spr  + (long)rn * KS, 0, 3);
      __builtin_prefetch(z    + (long)rn * KZ, 0, 3);
      __builtin_prefetch(zmix + (long)rn * KZ, 0, 3);
    }
    const int col = jt * 16 + (lane & 15);
    const float Rc  = Rpoint[col];
    const float Rec = Rema[col];
    v8f accf = mm_tile<KF>(Af,  f,    jt * 16);
    v8f accs = mm_tile<KS>(Asp, spr,  jt * 16);
    v8f accz = mm_tile<KZ>(Az,  z,    jt * 16);
    v8f accm = mm_tile<KZ>(Am,  zmix, jt * 16);
#pragma unroll
    for (int v = 0; v < 8; ++v) {
      const float sf = accf[v];
      const float ss = accs[v];
      const float lz = accz[v] * invT;
      const float lm = accm[v] * invT;
      const float neg = (rowg[v] != col) ? 1.f : 0.f;
      const float mf = ((sf > rr[v])  || (sf > Rc))  ? 1.f : 0.f;
      const float me = ((ss > rre[v]) || (ss > Rec)) ? 1.f : 0.f;
      const float u  = U[srow[v] * BTOT + col];
      const float nn = neg * (((mf != 0.f) || (u < RHO_F)) ? 1.f : 0.f);
      // logits bounded by 1/TEMP (unit vectors) -> exp() safe without rowmax
      dz[v] += nn * __expf(lz);
      dm[v] += nn * __expf(lm);
      const float w = neg * (mf + LMBD_F * me);
      msum[v] += neg * mf;
      wsum[v] += w;
      swz[v]  += w * lz;
      swm[v]  += w * lm;
    }
  }

  // Reduce each quantity over the 16 lanes holding one C-matrix row.
#pragma unroll
  for (int v = 0; v < 8; ++v) {
#pragma unroll
    for (int m = 8; m >= 1; m >>= 1) {
      msum[v] += __shfl_xor(msum[v], m, 32);
      wsum[v] += __shfl_xor(wsum[v], m, 32);
      swz[v]  += __shfl_xor(swz[v],  m, 32);
      swm[v]  += __shfl_xor(swm[v],  m, 32);
      dz[v]   += __shfl_xor(dz[v],   m, 32);
      dm[v]   += __shfl_xor(dm[v],   m, 32);
    }
    if ((lane & 15) == 0) {
      const int r = v + 8 * h;
      red[wave][r][0] = msum[v];
      red[wave][r][1] = wsum[v];
      red[wave][r][2] = swz[v];
      red[wave][r][3] = swm[v];
      red[wave][r][4] = dz[v];
      red[wave][r][5] = dm[v];
    }
  }
  __syncthreads();

  if (threadIdx.x < 16) {
    const int r = threadIdx.x;
    float q[6] = {0.f, 0.f, 0.f, 0.f, 0.f, 0.f};
#pragma unroll
    for (int w = 0; w < 8; ++w)
#pragma unroll
      for (int c = 0; c < 6; ++c) q[c] += red[w][r][c];
    float* p = part + ((long)(i * NSPL + sp) * 16 + r) * 6;
#pragma unroll
    for (int c = 0; c < 6; ++c) p[c] = q[c];
  }
}

// Merge column splits per row, then per-row finalize:
// rowval = valid*(swz+swm - wsum*(log dz + log dm)), wtot = valid*wsum.
__global__ void rowcombine(const float* __restrict__ part,
                           float* __restrict__ rowval,
                           float* __restrict__ wtot) {
  const int r = blockIdx.x * 256 + threadIdx.x;    // 0..4095
  const int i = r >> 4, lr = r & 15;
  float q[6] = {0.f, 0.f, 0.f, 0.f, 0.f, 0.f};
  for (int sp = 0; sp < NSPL; ++sp) {
    const float* p = part + ((long)(i * NSPL + sp) * 16 + lr) * 6;
#pragma unroll
    for (int c = 0; c < 6; ++c) q[c] += p[c];
  }
  const float valid = (q[0] != 0.f) ? 1.f : 0.f;
  float rv = 0.f;
  if (valid != 0.f) rv = q[2] + q[3] - q[1] * (__logf(q[4]) + __logf(q[5]));
  rowval[r] = rv;
  wtot[r]   = valid * q[1];
}

// Per-block (M=16 blocks of 256 rows) reduction -> scalar loss.
// n_valid cancels: block_loss = -scale * sum(rowval)/sum(wtot).
__global__ void supcon_finalize(const float* __restrict__ rowval,
                                const float* __restrict__ wtot,
                                float* __restrict__ out) {
  __shared__ float s1[256], s2[256];
  const int t = threadIdx.x;
  float loss = 0.f;
  for (int b = 0; b < MBLK; ++b) {
    s1[t] = rowval[b * SPLIT + t];
    s2[t] = wtot[b * SPLIT + t];
    __syncthreads();
    for (int off = 128; off > 0; off >>= 1) {
      if (t < off) { s1[t] += s1[t + off]; s2[t] += s2[t + off]; }
      __syncthreads();
    }
    if (t == 0) {
      const float w = (s2[0] > 0.f) ? s2[0] : 1.f;
      loss += s1[0] / w;
    }
    __syncthreads();
  }
  if (t == 0) {
    const float scale = TEMP_F / BASE_TEMP_F;
    out[0] = -scale * loss / ((float)MBLK * 2.0f);
  }
}

extern "C" void kernel_launch(void* const* d_in, const int* in_sizes, int n_in,
                              void* d_out, int out_size, void* d_ws, size_t ws_size,
                              hipStream_t stream) {
  const float* z    = (const float*)d_in[0];   // [4096,128]
  const float* spr  = (const float*)d_in[1];   // [4096,70]
  const float* f    = (const float*)d_in[2];   // [4096,384]
  const float* pag  = (const float*)d_in[3];   // [2048,384]
  const float* psp  = (const float*)d_in[4];   // [2048,70]
  const float* zmix = (const float*)d_in[5];   // [4096,128]
  const float* U    = (const float*)d_in[6];   // [256,4096]
  (void)in_sizes; (void)n_in; (void)out_size; (void)ws_size;

  float* ws     = (float*)d_ws;
  float* Rpoint = ws;                          // [4096]
  float* Rema   = ws + BTOT;                   // [4096]
  float* rowval = ws + 2 * BTOT;               // [4096]
  float* wtot   = ws + 3 * BTOT;               // [4096]
  float* part   = ws + 4 * BTOT;               // [256*NSPL*16*6] = 384KB

  rowmax_kernel<KF><<<BTOT / 16, 128, 0, stream>>>(f,   pag, Rpoint, NPOOL / 16);
  rowmax_kernel<KS><<<BTOT / 16, 128, 0, stream>>>(spr, psp, Rema,   NPOOL / 16);
  supcon_main<<<(BTOT / 16) * NSPL, 256, 0, stream>>>(f, spr, z, zmix, U,
                                                      Rpoint, Rema, part);
  rowcombine<<<BTOT / 256, 256, 0, stream>>>(part, rowval, wtot);
  supcon_finalize<<<1, 256, 0, stream>>>(rowval, wtot, (float*)d_out);
}